// Direction_15075335209585
// MI455X (gfx1250) — compile-verified
//
#include <hip/hip_runtime.h>

typedef float v2f __attribute__((ext_vector_type(2)));
typedef float v8f __attribute__((ext_vector_type(8)));

#define DDIM 26
#define ODIM 512
#define QS_STRIDE 28   // pad 26 -> 28 (zeroed) : K-padding + LDS bank spread

// ---------------------------------------------------------------------------
// Kernel 1: Householder QR (LAPACK sign convention) of W+1e-8, 512x26.
// One workgroup, 512 threads (16 waves of 32). Writes Q [512][26] row-major
// to global workspace.
// ---------------------------------------------------------------------------
__global__ __launch_bounds__(512) void qr512x26_kernel(const float* __restrict__ W,
                                                       float* __restrict__ Qg)
{
    __shared__ float As[ODIM][27];
    __shared__ float sbeta[DDIM];
    __shared__ float sdot[DDIM];

    const int tid  = threadIdx.x;
    const int lane = tid & 31;
    const int wave = tid >> 5;

    // Load A = W + 1e-8 (thread t owns row t)
    for (int j = 0; j < DDIM; ++j)
        As[tid][j] = W[tid * DDIM + j] + 1e-8f;
    __syncthreads();

    // ---- Phase 1: geqrf. After step k, column k of As holds v_k (rows >= k).
    for (int k = 0; k < DDIM; ++k) {
        if (wave == 0) {
            float n2 = 0.f;
            for (int t = lane; t < ODIM; t += 32)
                if (t >= k) { float x = As[t][k]; n2 += x * x; }
            for (int m = 16; m >= 1; m >>= 1) n2 += __shfl_xor(n2, m, 32);
            float x0    = As[k][k];
            float alpha = -copysignf(sqrtf(n2), x0);   // = R[k,k], LAPACK sign
            float v0    = x0 - alpha;
            float vv    = n2 - x0 * x0 + v0 * v0;
            float beta  = (vv > 0.f) ? 2.f / vv : 0.f;
            if (lane == 0) { sbeta[k] = beta; As[k][k] = v0; }
        }
        __syncthreads();
        const float beta = sbeta[k];
        // Each wave reduces full columns independently (no cross-wave reduce)
        for (int j = k + 1 + wave; j < DDIM; j += 16) {
            float s = 0.f;
            for (int t = lane; t < ODIM; t += 32)
                if (t >= k) s += As[t][k] * As[t][j];
            for (int m = 16; m >= 1; m >>= 1) s += __shfl_xor(s, m, 32);
            if (lane == 0) sdot[j] = s * beta;
        }
        __syncthreads();
        if (tid >= k) {
            const float vt = As[tid][k];
            for (int j = k + 1; j < DDIM; ++j)
                As[tid][j] -= sdot[j] * vt;
        }
        __syncthreads();
    }

    // ---- Phase 2: orgqr. Q[:,j] = H_0 ... H_25 e_j ; one column per wave,
    // column held in registers (16 f32 per lane), reflectors read from LDS.
    for (int j = wave; j < DDIM; j += 16) {
        float w[16];
        #pragma unroll
        for (int c = 0; c < 16; ++c) {
            int i = lane + 32 * c;
            w[c] = (i == j) ? 1.f : 0.f;
        }
        for (int k = DDIM - 1; k >= 0; --k) {
            const float beta = sbeta[k];
            float dot = 0.f;
            #pragma unroll
            for (int c = 0; c < 16; ++c) {
                int i = lane + 32 * c;
                if (i >= k) dot += As[i][k] * w[c];
            }
            for (int m = 16; m >= 1; m >>= 1) dot += __shfl_xor(dot, m, 32);
            const float t = beta * dot;
            #pragma unroll
            for (int c = 0; c < 16; ++c) {
                int i = lane + 32 * c;
                if (i >= k) w[c] -= t * As[i][k];
            }
        }
        #pragma unroll
        for (int c = 0; c < 16; ++c)
            Qg[(lane + 32 * c) * DDIM + j] = w[c];
    }
}

// ---------------------------------------------------------------------------
// Kernel 2: out[b,o] = sum_d X[b,d] * Q[o,d], via V_WMMA_F32_16X16X4_F32.
// 256 threads (8 waves); each wave owns a 16-row M tile and sweeps all 32
// 16-col N tiles, reusing its A fragments. Q staged in LDS (stride 28,
// zero-padded so the K=24..27 slice reads zeros).
// ---------------------------------------------------------------------------
__global__ __launch_bounds__(256) void dirproj_wmma_kernel(const float* __restrict__ X,
                                                           const float* __restrict__ Qg,
                                                           float* __restrict__ Y)
{
    __shared__ float Qs[ODIM * QS_STRIDE];

    const int tid  = threadIdx.x;
    const int lane = tid & 31;
    const int wave = tid >> 5;

    // Stage Q into LDS with padding
    for (int r = tid; r < ODIM; r += 256) {
        #pragma unroll
        for (int c = 0; c < DDIM; ++c)
            Qs[r * QS_STRIDE + c] = Qg[r * DDIM + c];
        Qs[r * QS_STRIDE + 26] = 0.f;
        Qs[r * QS_STRIDE + 27] = 0.f;
    }
    __syncthreads();

    const int m0  = (blockIdx.x * 8 + wave) * 16;
    const int mr  = lane & 15;          // row within tile (both halves)
    const int kh  = (lane >> 4) * 2;    // K sub-pair: 0 or 2

    // A fragments: 7 K-slices of 4 (K padded 26 -> 28)
    v2f a[7];
    const v2f zz = {0.f, 0.f};
    #pragma unroll
    for (int s = 0; s < 7; ++s) {
        const int k = 4 * s + kh;
        if (k + 1 < DDIM) {
            // (row*26 + k) is even -> 8-byte aligned float2 load
            a[s] = *reinterpret_cast<const v2f*>(X + (size_t)(m0 + mr) * DDIM + k);
        } else {
            a[s] = zz;  // k = 26,27 padding
        }
    }

    const float* qbase = &Qs[mr * QS_STRIDE + kh];

    #pragma unroll 1
    for (int nt = 0; nt < 32; ++nt) {
        const int n0 = nt * 16;
        v8f acc = {0.f, 0.f, 0.f, 0.f, 0.f, 0.f, 0.f, 0.f};
        const float* qb = qbase + n0 * QS_STRIDE;
        #pragma unroll
        for (int s = 0; s < 7; ++s) {
            v2f b = *reinterpret_cast<const v2f*>(qb + 4 * s);  // ds_load_b64
            acc = __builtin_amdgcn_wmma_f32_16x16x4_f32(
                /*neg_a=*/false, a[s], /*neg_b=*/false, b,
                /*c_mod=*/(short)0, acc, /*reuse_a=*/false, /*reuse_b=*/false);
        }
        // C layout: VGPR v -> rows (m0+v | m0+v+8), cols n0 + lane&15
        float* yp = Y + (size_t)(m0 + (lane >> 4) * 8) * ODIM + n0 + mr;
        #pragma unroll
        for (int v = 0; v < 8; ++v)
            __builtin_nontemporal_store(acc[v], yp + (size_t)v * ODIM);
    }
}

extern "C" void kernel_launch(void* const* d_in, const int* in_sizes, int n_in,
                              void* d_out, int out_size, void* d_ws, size_t ws_size,
                              hipStream_t stream) {
    const float* X = (const float*)d_in[0];   // [B, 26] f32
    const float* W = (const float*)d_in[1];   // [512, 26] f32
    float*       Y = (float*)d_out;           // [B, 512] f32
    float*       Qg = (float*)d_ws;           // [512, 26] f32 scratch

    const int B = in_sizes[0] / DDIM;         // 262144
    const int grid = B / 128;                 // 8 waves x 16 rows per block

    qr512x26_kernel<<<1, 512, 0, stream>>>(W, Qg);
    dirproj_wmma_kernel<<<grid, 256, 0, stream>>>(X, Qg, Y);
}